// GGNN_1726576856971
// MI455X (gfx1250) — compile-verified
//
#include <hip/hip_runtime.h>
#include <math.h>

typedef __attribute__((ext_vector_type(2))) float v2f;
typedef __attribute__((ext_vector_type(8))) float v8f;

#define FDIM     64
#define INTERNAL 128
#define CARD     32
#define EPW      16   // edges per wave (WMMA M)
#define EPB      64   // edges per block (4 waves)
#define S64      66   // padded LDS stride for 64-wide rows
#define S128     130  // padded LDS stride for 128-wide rows
#define SWB      66   // padded stride for bilinear weight chunks

// Branchless ELU: always evaluate expm1 on clamped input, select with cndmask.
__device__ __forceinline__ float eluf(float x) {
  const float en = expf(fminf(x, 0.f)) - 1.f;
  return x > 0.f ? x : en;
}

__device__ __forceinline__ v8f wmma4(v2f a, v2f b, v8f c) {
  // D(16x16,f32) = A(16x4,f32) * B(4x16,f32) + C
  return __builtin_amdgcn_wmma_f32_16x16x4_f32(
      /*neg_a=*/false, a, /*neg_b=*/false, b,
      /*c_mod=*/(short)0, c, /*reuse_a=*/false, /*reuse_b=*/false);
}

// O[16][ostride] = ELU( A[16][kdim] @ B[ntiles*16][kdim]^T )
// A rows: M=edge; B rows: output feature n, each row holds K contiguous.
// Dual accumulators over the two K halves -> two independent WMMA chains.
template <int KDIM, int NTILES>
__device__ __forceinline__ void proj_elu(const float* A, int astride,
                                         const float* B, int bstride,
                                         float* O, int ostride, int lane) {
  const int lo = lane & 15;
  const int hi = lane >> 4;  // 0: K={0,1}, 1: K={2,3} per ISA A-layout
  const int KH = KDIM / 2;
  const float* arow = A + lo * astride + 2 * hi;
#pragma unroll 1
  for (int nt = 0; nt < NTILES; ++nt) {
    const int n0 = nt << 4;
    const float* brow = B + (n0 + lo) * bstride + 2 * hi;
    v8f acc0 = {};
    v8f acc1 = {};
#pragma unroll
    for (int k0 = 0; k0 < KH; k0 += 4) {
      v2f a0 = *(const v2f*)(arow + k0);
      v2f b0 = *(const v2f*)(brow + k0);
      v2f a1 = *(const v2f*)(arow + KH + k0);
      v2f b1 = *(const v2f*)(brow + KH + k0);
      acc0 = wmma4(a0, b0, acc0);
      acc1 = wmma4(a1, b1, acc1);
    }
#pragma unroll
    for (int r = 0; r < 8; ++r) {  // C layout: VGPR r -> row r (+8 for hi half)
      O[(r + hi * 8) * ostride + n0 + lo] = eluf(acc0[r] + acc1[r]);
    }
  }
}

__global__ __launch_bounds__(128) void ggnn_edge_kernel(
    const float* __restrict__ nodef, const float* __restrict__ edgef,
    const int* __restrict__ esrc, const int* __restrict__ etgt,
    const float* __restrict__ W1_0, const float* __restrict__ W2_0,
    const float* __restrict__ Wb_0, const float* __restrict__ Wo_0,
    const float* __restrict__ W1_1, const float* __restrict__ W2_1,
    const float* __restrict__ Wb_1, const float* __restrict__ Wo_1,
    float* __restrict__ agg, int E) {
  __shared__ float s_w[INTERNAL * S64];  // 8448 f: W1/W2 as [128][S64], Wo as [64][S128]
  __shared__ float s_wb[CARD * SWB];     // 2112 f
  __shared__ float s_h[EPB * S64];       // 4224 f
  __shared__ float s_z[EPB * S64];       // 4224 f
  __shared__ float s_x1[EPB * S128];     // 8320 f
  __shared__ float s_x2[EPB * S128];     // 8320 f

  const int tid  = threadIdx.x;
  const int wave = tid >> 5;
  const int lane = tid & 31;
  const int e0   = blockIdx.x * EPB;

  // Gather source-node rows (L2-resident table) + edge feature rows.
  for (int i = tid; i < EPB * FDIM; i += 128) {
    const int e = i >> 6, f = i & 63;
    int ge = e0 + e;
    if (ge >= E) ge = E - 1;
    s_h[e * S64 + f] = nodef[esrc[ge] * FDIM + f];
    s_z[e * S64 + f] = edgef[ge * FDIM + f];
  }

  const float* hA  = s_h  + wave * EPW * S64;
  float*       zW  = s_z  + wave * EPW * S64;
  float*       x1W = s_x1 + wave * EPW * S128;
  float*       x2W = s_x2 + wave * EPW * S128;

#pragma unroll 1
  for (int layer = 0; layer < 2; ++layer) {
    const float* W1 = layer ? W1_1 : W1_0;
    const float* W2 = layer ? W2_1 : W2_0;
    const float* Wb = layer ? Wb_1 : Wb_0;
    const float* Wo = layer ? Wo_1 : Wo_0;

    // x1 = ELU(h @ W1^T)
    __syncthreads();
    for (int i = tid; i < INTERNAL * FDIM; i += 128)
      s_w[(i >> 6) * S64 + (i & 63)] = W1[i];
    __syncthreads();
    proj_elu<FDIM, INTERNAL / 16>(hA, S64, s_w, S64, x1W, S128, lane);

    // x2 = ELU(z @ W2^T)
    __syncthreads();
    for (int i = tid; i < INTERNAL * FDIM; i += 128)
      s_w[(i >> 6) * S64 + (i & 63)] = W2[i];
    __syncthreads();
    proj_elu<FDIM, INTERNAL / 16>(zW, S64, s_w, S64, x2W, S128, lane);

    // Stage Wo [64][128] and Wb [32][4][4][4]
    __syncthreads();
    for (int i = tid; i < FDIM * INTERNAL; i += 128)
      s_w[(i >> 7) * S128 + (i & 127)] = Wo[i];
    for (int i = tid; i < CARD * 64; i += 128)
      s_wb[(i >> 6) * SWB + (i & 63)] = Wb[i];
    __syncthreads();

    // Chunked bilinear: y[e,c,o] = ELU(sum_ij x1[e,c,i]*Wb[c,o,i,j]*x2[e,c,j])
    // lane == chunk c; overwrite x1W in place (each lane owns its slot).
#pragma unroll 1
    for (int e = 0; e < EPW; ++e) {
      const int c = lane;
      float* xa = x1W + e * S128 + c * 4;
      const float* xb = x2W + e * S128 + c * 4;
      float a[4], b[4];
#pragma unroll
      for (int i = 0; i < 4; ++i) { a[i] = xa[i]; b[i] = xb[i]; }
      const float* wb = s_wb + c * SWB;
      float y[4];
#pragma unroll
      for (int o = 0; o < 4; ++o) {
        float acc = 0.f;
#pragma unroll
        for (int i2 = 0; i2 < 4; ++i2)
#pragma unroll
          for (int j = 0; j < 4; ++j)
            acc = fmaf(a[i2] * b[j], wb[o * 16 + i2 * 4 + j], acc);
        y[o] = eluf(acc);
      }
#pragma unroll
      for (int o = 0; o < 4; ++o) xa[o] = y[o];
    }

    // z = ELU(y @ Wo^T)  (same-wave LDS ops are in-order; no barrier needed)
    proj_elu<INTERNAL, FDIM / 16>(x1W, S128, s_w, S128, zW, S64, lane);
  }

  // Scatter-add messages to target nodes (native global_atomic_add_f32).
  __syncthreads();
  for (int i = tid; i < EPB * FDIM; i += 128) {
    const int e = i >> 6, f = i & 63;
    const int ge = e0 + e;
    if (ge < E)
      unsafeAtomicAdd(&agg[etgt[ge] * FDIM + f], s_z[e * S64 + f]);
  }
}

__global__ __launch_bounds__(256) void ggnn_gru_kernel(
    const float* __restrict__ nodef, const float* __restrict__ agg,
    const float* __restrict__ Wih, const float* __restrict__ Whh,
    const float* __restrict__ bih, const float* __restrict__ bhh,
    float* __restrict__ out, int N) {
  __shared__ float sa[4][FDIM];
  __shared__ float sh[4][FDIM];
  const int g = threadIdx.x >> 6;  // node slot within block
  const int f = threadIdx.x & 63;
  const int node = blockIdx.x * 4 + g;
  if (node < N) {
    sa[g][f] = agg[node * FDIM + f];
    sh[g][f] = nodef[node * FDIM + f];
  }
  __syncthreads();
  if (node >= N) return;

  float ir = bih[f], iz = bih[64 + f], in_ = bih[128 + f];
  float hr = bhh[f], hz = bhh[64 + f], hn = bhh[128 + f];
  const float* wr = Wih + f * FDIM;
  const float* wz = Wih + (64 + f) * FDIM;
  const float* wn = Wih + (128 + f) * FDIM;
  const float* vr = Whh + f * FDIM;
  const float* vz = Whh + (64 + f) * FDIM;
  const float* vn = Whh + (128 + f) * FDIM;
#pragma unroll 4
  for (int k = 0; k < FDIM; ++k) {
    const float a = sa[g][k], h = sh[g][k];
    ir = fmaf(a, wr[k], ir); iz = fmaf(a, wz[k], iz); in_ = fmaf(a, wn[k], in_);
    hr = fmaf(h, vr[k], hr); hz = fmaf(h, vz[k], hz); hn  = fmaf(h, vn[k], hn);
  }
  const float r = 1.f / (1.f + expf(-(ir + hr)));
  const float u = 1.f / (1.f + expf(-(iz + hz)));
  const float n = tanhf(in_ + r * hn);
  out[node * FDIM + f] = (1.f - u) * n + u * sh[g][f];
}

__global__ void ggnn_zero_kernel(float* __restrict__ p, int n) {
  const int i = blockIdx.x * 256 + threadIdx.x;
  if (i < n) p[i] = 0.f;
}

extern "C" void kernel_launch(void* const* d_in, const int* in_sizes, int n_in,
                              void* d_out, int out_size, void* d_ws, size_t ws_size,
                              hipStream_t stream) {
  const float* nodef = (const float*)d_in[0];
  const float* edgef = (const float*)d_in[1];
  const int*   esrc  = (const int*)d_in[2];
  const int*   etgt  = (const int*)d_in[3];
  const float* W1_0  = (const float*)d_in[4];
  const float* W2_0  = (const float*)d_in[5];
  const float* Wb_0  = (const float*)d_in[6];
  const float* Wo_0  = (const float*)d_in[7];
  const float* W1_1  = (const float*)d_in[8];
  const float* W2_1  = (const float*)d_in[9];
  const float* Wb_1  = (const float*)d_in[10];
  const float* Wo_1  = (const float*)d_in[11];
  const float* Wih   = (const float*)d_in[12];
  const float* Whh   = (const float*)d_in[13];
  const float* bih   = (const float*)d_in[14];
  const float* bhh   = (const float*)d_in[15];

  const int N = in_sizes[0] / FDIM;
  const int E = in_sizes[2];
  float* agg = (float*)d_ws;  // N*FDIM floats of scratch
  float* out = (float*)d_out;

  ggnn_zero_kernel<<<(N * FDIM + 255) / 256, 256, 0, stream>>>(agg, N * FDIM);
  ggnn_edge_kernel<<<(E + EPB - 1) / EPB, 128, 0, stream>>>(
      nodef, edgef, esrc, etgt,
      W1_0, W2_0, Wb_0, Wo_0, W1_1, W2_1, Wb_1, Wo_1, agg, E);
  ggnn_gru_kernel<<<(N + 3) / 4, 256, 0, stream>>>(
      nodef, agg, Wih, Whh, bih, bhh, out, N);
}